// BesselFunction_41798621724976
// MI455X (gfx1250) — compile-verified
//
#include <hip/hip_runtime.h>
#include <hip/hip_bf16.h>
#include <math.h>

// Elementwise Bessel J2 over 67.1M fp32 elements.
// Memory-bound target: 537 MB streamed / 23.3 TB/s ~= 23 us. Both input and
// output exceed/compete with the 192 MB L2, so use non-temporal (NT) 128-bit
// vector loads/stores. VALU side is trimmed to stay under the roof:
//  - hardware v_sin/v_cos once, shifted phases via angle-addition FMAs
//  - v_rcp_f32 for all reciprocals (no IEEE divide expansion)
//  - v_rsq_f32 for the asymptotic amplitude sqrt(2/(pi*ax))
//  - fully branchless (wave32 lanes mix |x|<8 and |x|>=8 in almost every wave)
// Grid is sized so each thread runs ~8 grid-stride iterations: the ~140-op
// coefficient/SGPR setup the compiler hoists out of the loop is amortized
// across 32 elements instead of 4, and the loop lets the scheduler prefetch
// the next NT b128 load above the current iteration's VALU body.

#define INV_TWO_PI   0.15915494309189535f   // 1/(2*pi)
#define INV_SQRT2    0.70710678118654752f   // 1/sqrt(2)
#define SQRT_2_PI    0.79788456080286536f   // sqrt(2/pi)

// Native clang vector type: required by __builtin_nontemporal_load/store
// (HIP's float4 is a class and is rejected by the builtin).
typedef float v4f __attribute__((ext_vector_type(4)));

__device__ __forceinline__ float bessj2_f32(float x) {
    const float ax = __builtin_fabsf(x);
    const float y  = x * x;

    // ---- |x| < 8 : rational approximations (A&S 9.4.x / NR) ----
    float num0 = fmaf(y, fmaf(y, fmaf(y, fmaf(y, fmaf(y, -184.9052456f,
                        77392.33017f), -11214424.18f), 651619640.7f),
                        -13362590354.0f), 57568490574.0f);
    float den0 = fmaf(y, fmaf(y, fmaf(y, fmaf(y, (y + 267.8532712f),
                        59272.64853f), 9494680.718f), 1029532985.0f),
                        57568490411.0f);
    const float j0_small = num0 * __builtin_amdgcn_rcpf(den0);

    float num1 = x * fmaf(y, fmaf(y, fmaf(y, fmaf(y, fmaf(y, -30.16036606f,
                        15704.48260f), -2972611.439f), 242396853.1f),
                        -7895059235.0f), 72362614232.0f);
    float den1 = fmaf(y, fmaf(y, fmaf(y, fmaf(y, (y + 376.9991397f),
                        99447.43394f), 18583304.74f), 2300535178.0f),
                        144725228442.0f);
    const float j1_small = num1 * __builtin_amdgcn_rcpf(den1);

    // ---- |x| >= 8 : asymptotic expansion ----
    const float ax_s = __builtin_fmaxf(ax, 1e-20f);
    const float rax  = __builtin_amdgcn_rcpf(ax_s);     // 1/ax
    const float z    = 8.0f * rax;
    const float yy   = z * z;

    float p1_0 = fmaf(yy, fmaf(yy, fmaf(yy, fmaf(yy, 0.2093887211e-6f,
                        -0.2073370639e-5f), 0.2734510407e-4f),
                        -0.1098628627e-2f), 1.0f);
    float p2_0 = fmaf(yy, fmaf(yy, fmaf(yy, fmaf(yy, -0.934935152e-7f,
                        0.7621095161e-6f), -0.6911147651e-5f),
                        0.1430488765e-3f), -0.1562499995e-1f);
    float p1_1 = fmaf(yy, fmaf(yy, fmaf(yy, fmaf(yy, -0.240337019e-6f,
                        0.2457520174e-5f), -0.3516396496e-4f),
                        0.183105e-2f), 1.0f);
    float p2_1 = fmaf(yy, fmaf(yy, fmaf(yy, fmaf(yy, 0.105787412e-6f,
                        -0.88228987e-6f), 0.8449199096e-5f),
                        -0.2002690873e-3f), 0.04687499995f);

    // sin/cos of ax ONCE (v_sin_f32/v_cos_f32 take revolutions), then derive
    //   cos(ax - pi/4)  = (c+s)/sqrt2     sin(ax - pi/4)  = (s-c)/sqrt2
    //   cos(ax - 3pi/4) = (s-c)/sqrt2     sin(ax - 3pi/4) = -(c+s)/sqrt2
    const float rev = ax_s * INV_TWO_PI;
    const float s   = __builtin_amdgcn_sinf(rev);
    const float c   = __builtin_amdgcn_cosf(rev);
    const float cps = (c + s) * INV_SQRT2;   // cos(ax-pi/4)
    const float smc = (s - c) * INV_SQRT2;   // sin(ax-pi/4) == cos(ax-3pi/4)

    const float amp = SQRT_2_PI * __builtin_amdgcn_rsqf(ax_s); // sqrt(2/(pi*ax))

    // j0_big = amp*(cos(xx0)*p1_0 - z*sin(xx0)*p2_0)
    const float j0_big = amp * fmaf(-z * smc, p2_0, cps * p1_0);
    // j1_big = sign(x)*amp*(cos(xx1)*p1_1 - z*sin(xx1)*p2_1)
    //        = sign(x)*amp*(smc*p1_1 + z*cps*p2_1)
    const float j1_big = __builtin_copysignf(
        amp * fmaf(z * cps, p2_1, smc * p1_1), x);

    const bool small_arg = ax < 8.0f;
    const float j0 = small_arg ? j0_small : j0_big;
    const float j1 = small_arg ? j1_small : j1_big;

    // J2 = (2/x) J1 - J0 ; near 0 use the series J2 ~ x^2/8.
    // Reuse rcp(ax) with copysign instead of a second v_rcp.
    const float rx  = __builtin_copysignf(rax, x);
    const float rec = fmaf(2.0f * rx, j1, -j0);
    const float ser = y * 0.125f;
    return (ax < 1e-4f) ? ser : rec;
}

__global__ __launch_bounds__(256)
void BesselFunction_41798621724976_kernel(const float* __restrict__ x,
                                          float* __restrict__ out,
                                          int n4, int n) {
    const int tid    = blockIdx.x * blockDim.x + threadIdx.x;
    const int stride = gridDim.x * blockDim.x;

    const v4f* __restrict__ x4 = reinterpret_cast<const v4f*>(x);
    v4f*       __restrict__ o4 = reinterpret_cast<v4f*>(out);

    // Main body: one float4 per thread per iteration, NT 128-bit traffic.
    // ~8 iterations per thread amortize the hoisted coefficient setup.
    for (int i = tid; i < n4; i += stride) {
        v4f v = __builtin_nontemporal_load(&x4[i]);
        v4f r;
        r.x = bessj2_f32(v.x);
        r.y = bessj2_f32(v.y);
        r.z = bessj2_f32(v.z);
        r.w = bessj2_f32(v.w);
        __builtin_nontemporal_store(r, &o4[i]);
    }

    // Scalar tail (n not a multiple of 4 — empty for the reference shape).
    const int base = n4 * 4;
    const int t    = base + tid;
    if (t < n) {
        out[t] = bessj2_f32(__builtin_nontemporal_load(&x[t]));
    }
}

extern "C" void kernel_launch(void* const* d_in, const int* in_sizes, int n_in,
                              void* d_out, int out_size, void* d_ws, size_t ws_size,
                              hipStream_t stream) {
    (void)n_in; (void)d_ws; (void)ws_size; (void)out_size;

    const float* x   = reinterpret_cast<const float*>(d_in[0]);
    float*       out = reinterpret_cast<float*>(d_out);
    const int n  = in_sizes[0];
    const int n4 = n >> 2;

    const int block = 256;                 // 8 wave32 waves per block
    // Cap the grid so each thread runs ~8 grid-stride iterations
    // (16.8M float4s / (8192*256) = 8): amortizes per-thread constant setup
    // while 65536 waves still oversubscribe every WGP.
    int grid = (n4 + block - 1) / block;
    if (grid < 1)    grid = 1;
    if (grid > 8192) grid = 8192;

    BesselFunction_41798621724976_kernel<<<grid, block, 0, stream>>>(x, out, n4, n);
}